// GraphSAGE_61529701482749
// MI455X (gfx1250) — compile-verified
//
#include <hip/hip_runtime.h>
#include <hip/hip_bf16.h>

// ---------------------------------------------------------------------------
// GraphSAGE (2-layer mean SAGEConv) for MI455X / gfx1250, wave32.
//   h1  = X @ Wself1 + (segmean X) @ Wneigh1 + b1
//   out = h1 @ Wself2 + (segmean h1) @ Wneigh2 + b2
// Dense parts use V_WMMA_F32_16X16X4_F32 (fp32 WMMA, keeps reference precision;
// GEMM is ~2.5 GFLOP, totally dwarfed by the ~1.2 GB of edge traffic).
// ---------------------------------------------------------------------------

typedef __attribute__((ext_vector_type(2))) float v2f;
typedef __attribute__((ext_vector_type(8))) float v8f;

#define N_NODES 100000
#define FEAT    64          // IN == HID == 64

// ---------------------------------------------------------------- zero fill
__global__ void zero_kernel(float* __restrict__ p, int count) {
    int i = blockIdx.x * blockDim.x + threadIdx.x;
    if (i < count) p[i] = 0.0f;
}

// ------------------------------------------------- edge scatter (+ degree)
// 16 threads per edge; each thread moves one float4 of the 64-float row.
__global__ void scatter_kernel(const float* __restrict__ X,   // [N,64]
                               const int*   __restrict__ src,
                               const int*   __restrict__ dst,
                               float* __restrict__ agg,       // [N,64]
                               float* __restrict__ deg,       // [N] or null
                               int n_edges) {
    int tid = blockIdx.x * blockDim.x + threadIdx.x;
    int total = n_edges * 16;
    if (tid >= total) return;
    int e = tid >> 4;
    int q = tid & 15;
    int s = src[e];
    int d = dst[e];
    const float4 v = ((const float4*)X)[s * 16 + q];
    float* p = agg + (size_t)d * FEAT + q * 4;
    atomicAdd(p + 0, v.x);
    atomicAdd(p + 1, v.y);
    atomicAdd(p + 2, v.z);
    atomicAdd(p + 3, v.w);
    if (deg != nullptr && q == 0) atomicAdd(deg + d, 1.0f);
}

// ------------------------------------------------------------- 1/max(deg,1)
__global__ void invdeg_kernel(float* __restrict__ deg, int n) {
    int i = blockIdx.x * blockDim.x + threadIdx.x;
    if (i < n) deg[i] = 1.0f / fmaxf(deg[i], 1.0f);
}

// --------------------------------------------------------------- WMMA GEMM
// Y[16-row tile] = X_tile @ Wself + (AGG_tile * inv) @ Wneigh + bias
// One wave per 16x16 output tile; NCOL/16 waves per block.
// A tiles staged in LDS, row stride 68 floats: rows stay 16B-aligned and the
// fragment column reads hit banks (4m+k)%64 -> conflict-free across m=0..15.
template <int NCOL>
__global__ __launch_bounds__(NCOL * 2) void sage_gemm_kernel(
        const float* __restrict__ X,      // [N,64]
        const float* __restrict__ AGG,    // [N,64] neighbor sums
        const float* __restrict__ INV,    // [N]    1/max(deg,1)
        const float* __restrict__ Wself,  // [64,NCOL] row-major
        const float* __restrict__ Wneigh, // [64,NCOL]
        const float* __restrict__ bias,   // [NCOL]
        float* __restrict__ Y) {          // [N,NCOL]
    constexpr int LDSS = 68;
    __shared__ float ldsX[16 * LDSS];
    __shared__ float ldsA[16 * LDSS];

    const int row0 = blockIdx.x * 16;
    const int tid  = threadIdx.x;
    const int lane = tid & 31;
    const int wave = tid >> 5;

    // Cooperative, coalesced tile load; scale AGG by inv-degree on the way in.
    for (int idx = tid; idx < 16 * FEAT; idx += NCOL * 2) {
        int r = idx >> 6;
        int c = idx & 63;
        ldsX[r * LDSS + c] = X[(size_t)(row0 + r) * FEAT + c];
        ldsA[r * LDSS + c] = AGG[(size_t)(row0 + r) * FEAT + c] * INV[row0 + r];
    }
    __syncthreads();

    const int m    = lane & 15;          // A row / C col index
    const int hi   = lane >> 4;          // half-wave select
    const int kh   = hi * 2;             // K sub-offset per ISA f32 A layout
    const int col0 = wave * 16;

    // C preloaded with bias (per-column, identical across the 8 row slots).
    const float bv = bias[col0 + m];
    v8f acc;
#pragma unroll
    for (int i = 0; i < 8; ++i) acc[i] = bv;

#pragma unroll
    for (int k0 = 0; k0 < FEAT; k0 += 4) {
        const int kb = k0 + kh;
        v2f a, b;
        // A 16x4 f32: lanes 0-15 -> K=k0,k0+1 ; lanes 16-31 -> K=k0+2,k0+3
        a.x = ldsX[m * LDSS + kb];
        a.y = ldsX[m * LDSS + kb + 1];
        // B 4x16 f32: N striped over lanes, same K split across half-waves
        b.x = Wself[(size_t)kb * NCOL + col0 + m];
        b.y = Wself[(size_t)(kb + 1) * NCOL + col0 + m];
        acc = __builtin_amdgcn_wmma_f32_16x16x4_f32(false, a, false, b,
                                                    (short)0, acc, false, false);
        a.x = ldsA[m * LDSS + kb];
        a.y = ldsA[m * LDSS + kb + 1];
        b.x = Wneigh[(size_t)kb * NCOL + col0 + m];
        b.y = Wneigh[(size_t)(kb + 1) * NCOL + col0 + m];
        acc = __builtin_amdgcn_wmma_f32_16x16x4_f32(false, a, false, b,
                                                    (short)0, acc, false, false);
    }

    // C/D f32 16x16: VGPR i holds M=i (lanes 0-15) / M=8+i (lanes 16-31).
    const int mbase = hi * 8;
#pragma unroll
    for (int i = 0; i < 8; ++i)
        Y[(size_t)(row0 + mbase + i) * NCOL + col0 + m] = acc[i];
}

// ---------------------------------------------------------------------------
extern "C" void kernel_launch(void* const* d_in, const int* in_sizes, int n_in,
                              void* d_out, int out_size, void* d_ws, size_t ws_size,
                              hipStream_t stream) {
    const float* feat    = (const float*)d_in[0];
    const int*   src     = (const int*)d_in[1];
    const int*   dst     = (const int*)d_in[2];
    const float* Wself1  = (const float*)d_in[3];
    const float* Wneigh1 = (const float*)d_in[4];
    const float* b1      = (const float*)d_in[5];
    const float* Wself2  = (const float*)d_in[6];
    const float* Wneigh2 = (const float*)d_in[7];
    const float* b2      = (const float*)d_in[8];
    float*       out     = (float*)d_out;

    const int n_edges = in_sizes[1];
    const int n_nodes = N_NODES;

    // Workspace layout (floats): h1[N*64] | agg[N*64] | deg/inv[N]
    float* h1  = (float*)d_ws;
    float* agg = h1 + (size_t)n_nodes * FEAT;
    float* deg = agg + (size_t)n_nodes * FEAT;

    const int zcount1 = n_nodes * FEAT + n_nodes;   // agg + deg (contiguous)
    const int sthreads = n_edges * 16;
    const int row_tiles = n_nodes / 16;             // 100000/16 == 6250 exact

    // ---- layer 1 ----
    zero_kernel<<<(zcount1 + 255) / 256, 256, 0, stream>>>(agg, zcount1);
    scatter_kernel<<<(sthreads + 255) / 256, 256, 0, stream>>>(
        feat, src, dst, agg, deg, n_edges);
    invdeg_kernel<<<(n_nodes + 255) / 256, 256, 0, stream>>>(deg, n_nodes);
    sage_gemm_kernel<64><<<row_tiles, 128, 0, stream>>>(
        feat, agg, deg, Wself1, Wneigh1, b1, h1);

    // ---- layer 2 ----
    const int zcount2 = n_nodes * FEAT;
    zero_kernel<<<(zcount2 + 255) / 256, 256, 0, stream>>>(agg, zcount2);
    scatter_kernel<<<(sthreads + 255) / 256, 256, 0, stream>>>(
        h1, src, dst, agg, nullptr, n_edges);
    sage_gemm_kernel<32><<<row_tiles, 64, 0, stream>>>(
        h1, agg, deg, Wself2, Wneigh2, b2, out);
}